// Voxelizer_10866267259091
// MI455X (gfx1250) — compile-verified
//
#include <hip/hip_runtime.h>

typedef __attribute__((ext_vector_type(2))) float v2f;
typedef __attribute__((ext_vector_type(8))) float v8f;

#define NZg 128
#define NXg 128
#define NYg 128
#define MG  1024            // number of gaussians (K of the GEMM)
#define CUT2 9.0f           // SIGMA_CUTOFF^2
#define EPSg 1e-8f

// ---------------------------------------------------------------------------
// Kernel 1: build separable factor tables in workspace.
//   Wx[c][x], Wy[c][y]        (c-major; coordinate contiguous -> coalesced
//                              16-lane reads in the GEMM kernel)
//   Gr[z][c] = rho_r[c]*wz,   Gi[z][c] = rho_i[c]*wz
//                              (z-major; c contiguous -> packed float2 reads)
// Per-axis 3-sigma cutoff baked into the tables (box support; see analysis).
// ---------------------------------------------------------------------------
__global__ void voxelize_factors(const float* __restrict__ centers,
                                 const float* __restrict__ scales,
                                 const float* __restrict__ rho_r,
                                 const float* __restrict__ rho_i,
                                 float* __restrict__ Wx,
                                 float* __restrict__ Wy,
                                 float* __restrict__ Gr,
                                 float* __restrict__ Gi) {
  const int c = blockIdx.x;        // gaussian index
  const int i = threadIdx.x;       // coordinate index 0..127
  const float coord = -1.0f + 2.0f * (float)i * (1.0f / 127.0f);

  const float cz = centers[c * 3 + 0];
  const float cx = centers[c * 3 + 1];
  const float cy = centers[c * 3 + 2];
  const float sz = scales[c * 3 + 0] + EPSg;
  const float sx = scales[c * 3 + 1] + EPSg;
  const float sy = scales[c * 3 + 2] + EPSg;

  const float dz = (coord - cz) / sz;  const float dz2 = dz * dz;
  const float dx = (coord - cx) / sx;  const float dx2 = dx * dx;
  const float dy = (coord - cy) / sy;  const float dy2 = dy * dy;

  const float wz = (dz2 <= CUT2) ? expf(-0.5f * dz2) : 0.0f;
  const float wx = (dx2 <= CUT2) ? expf(-0.5f * dx2) : 0.0f;
  const float wy = (dy2 <= CUT2) ? expf(-0.5f * dy2) : 0.0f;

  Wx[c * NXg + i] = wx;
  Wy[c * NYg + i] = wy;
  Gr[i * MG + c] = rho_r[c] * wz;    // i plays the role of z here
  Gi[i * MG + c] = rho_i[c] * wz;
}

// ---------------------------------------------------------------------------
// Kernel 2: per-z-slab GEMM on the WMMA pipe.
//   vol[z,x,y] = sum_c (Wx[c,x]*G[z,c]) * Wy[c,y]   for real & imag channels
// One block per z (8 waves). Wave w owns x-tile-row w; accumulates all 8
// y-tiles x 2 channels in registers across the K=1024 loop (step 4).
//
// Fragment layouts (wave32, V_WMMA_F32_16X16X4_F32):
//   A 16x4 : lane -> row m=lane&15, K-pair {2*half, 2*half+1} (half=lane>>4)
//   B 4x16 : lane -> col n=lane&15, same K-pair per half
//   C 16x16: vgpr v, lane -> element (v + 8*half, lane&15)
// ---------------------------------------------------------------------------
__global__ void __launch_bounds__(256)
voxelize_wmma(const float* __restrict__ Wx,
              const float* __restrict__ Wy,
              const float* __restrict__ Gr,
              const float* __restrict__ Gi,
              float* __restrict__ out /* interleaved complex64 */) {
  const int z    = blockIdx.x;
  const int wave = threadIdx.x >> 5;   // x tile row, 0..7
  const int lane = threadIdx.x & 31;
  const int lo   = lane & 15;
  const int half = lane >> 4;

  const int xA = wave * 16 + lo;       // A-matrix row handled by this lane

  v8f accR[8];
  v8f accI[8];
#pragma unroll
  for (int t = 0; t < 8; ++t) {
    accR[t] = (v8f){0.f, 0.f, 0.f, 0.f, 0.f, 0.f, 0.f, 0.f};
    accI[t] = (v8f){0.f, 0.f, 0.f, 0.f, 0.f, 0.f, 0.f, 0.f};
  }

  const float2* __restrict__ GrP = (const float2*)(Gr + (size_t)z * MG);
  const float2* __restrict__ GiP = (const float2*)(Gi + (size_t)z * MG);

  for (int k0 = 0; k0 < MG; k0 += 4) {
    const int c0 = k0 + 2 * half;      // this lane's first K index (even)

    // A fragments for both channels (shared Wx, channel-specific G scale)
    const float2 gr = GrP[c0 >> 1];    // {Gr[z,c0], Gr[z,c0+1]}
    const float2 gi = GiP[c0 >> 1];
    const float wx0 = Wx[c0 * NXg + xA];
    const float wx1 = Wx[(c0 + 1) * NXg + xA];
    const v2f aR = {wx0 * gr.x, wx1 * gr.y};
    const v2f aI = {wx0 * gi.x, wx1 * gi.y};

#pragma unroll
    for (int ty = 0; ty < 8; ++ty) {
      const int y = ty * 16 + lo;
      const v2f b = {Wy[c0 * NYg + y], Wy[(c0 + 1) * NYg + y]};
      accR[ty] = __builtin_amdgcn_wmma_f32_16x16x4_f32(
          false, aR, false, b, (short)0, accR[ty], false, false);
      accI[ty] = __builtin_amdgcn_wmma_f32_16x16x4_f32(
          false, aI, false, b, (short)0, accI[ty], false, false);
    }
  }

  // Store: every voxel written exactly once, interleaved (re, im) = complex64.
  float2* __restrict__ outc = (float2*)out;
#pragma unroll
  for (int ty = 0; ty < 8; ++ty) {
#pragma unroll
    for (int v = 0; v < 8; ++v) {
      const int xr = wave * 16 + v + 8 * half;  // C row -> global x
      const int yc = ty * 16 + lo;              // C col -> global y
      const size_t idx = ((size_t)z * NXg + xr) * NYg + yc;
      outc[idx] = make_float2(accR[ty][v], accI[ty][v]);
    }
  }
}

// ---------------------------------------------------------------------------
// Workspace layout (floats): Wx | Wy | Gr | Gi  = 4 * 1024*128 * 4B = 2 MB
// ---------------------------------------------------------------------------
extern "C" void kernel_launch(void* const* d_in, const int* in_sizes, int n_in,
                              void* d_out, int out_size, void* d_ws, size_t ws_size,
                              hipStream_t stream) {
  const float* centers = (const float*)d_in[0];  // (1024, 3)
  const float* scales  = (const float*)d_in[1];  // (1024, 3)
  const float* rho_r   = (const float*)d_in[2];  // (1024,)
  const float* rho_i   = (const float*)d_in[3];  // (1024,)

  float* ws = (float*)d_ws;
  float* Wx = ws;
  float* Wy = ws + (size_t)MG * NXg;
  float* Gr = ws + 2 * (size_t)MG * NXg;
  float* Gi = ws + 3 * (size_t)MG * NXg;

  voxelize_factors<<<MG, 128, 0, stream>>>(centers, scales, rho_r, rho_i,
                                           Wx, Wy, Gr, Gi);
  voxelize_wmma<<<NZg, 256, 0, stream>>>(Wx, Wy, Gr, Gi, (float*)d_out);
}